// DegradationModel_81140522156584
// MI455X (gfx1250) — compile-verified
//
#include <hip/hip_runtime.h>
#include <hip/hip_bf16.h>

typedef __attribute__((ext_vector_type(2))) float v2f;
typedef __attribute__((ext_vector_type(4))) float v4f;
typedef __attribute__((ext_vector_type(8))) float v8f;

#define STRIDE 68  // 16-row LDS tile stride; bank-conflict-free, >= fin_max+1

static __device__ __forceinline__ v8f wmma4(v2f a, v2f b, v8f c) {
  // V_WMMA_F32_16X16X4_F32 : D[16x16] = A[16x4] * B[4x16] + C
  return __builtin_amdgcn_wmma_f32_16x16x4_f32(false, a, false, b, (short)0, c, false, false);
}

// ---------------------------------------------------------------------------
// Zero-fill of the (mostly zero) Hessian outputs: 277MB -> NT b128 stores.
// ---------------------------------------------------------------------------
__global__ void __launch_bounds__(256)
zero_f4_kernel(v4f* __restrict__ p, unsigned long long n4) {
  unsigned long long i = (unsigned long long)blockIdx.x * blockDim.x + threadIdx.x;
  if (i < n4) {
    v4f z = {0.0f, 0.0f, 0.0f, 0.0f};
    __builtin_nontemporal_store(z, p + i);
  }
}

// ---------------------------------------------------------------------------
// mean = dict_kernel[idx, :64], log_sig = dict_kernel[idx, 64:]
// ---------------------------------------------------------------------------
__global__ void __launch_bounds__(256)
gather_kernel(const float* __restrict__ dk, const int* __restrict__ idx,
              float* __restrict__ mean, float* __restrict__ logsig, int total) {
  int t = blockIdx.x * blockDim.x + threadIdx.x;
  if (t < total) {
    int n = t >> 6, j = t & 63;
    int id = idx[n];
    mean[t]   = dk[id * 128 + j];
    logsig[t] = dk[id * 128 + 64 + j];
  }
}

// ---------------------------------------------------------------------------
// One wave per 16-sample tile; all GEMMs via V_WMMA_F32_16X16X4_F32.
// FIN (67 cap / 66 vol) and IS_CAP are compile-time -> guard-free hot loops.
// ---------------------------------------------------------------------------
template <int FIN, bool IS_CAP>
__global__ void __launch_bounds__(32)
mlp_hess_kernel(const float* __restrict__ centers,   // N x 3
                const int*   __restrict__ idxs,      // N
                const float* __restrict__ measured,  // N
                const float* __restrict__ volts,     // V
                const float* __restrict__ eps,       // K x 64
                const float* __restrict__ dk,        // K x 128
                const float* __restrict__ w0,        // FIN x 64 (row-major)
                const float* __restrict__ b0,        // 64
                const float* __restrict__ wb,        // 3 x 64 x 64
                const float* __restrict__ bb,        // 3 x 64
                const float* __restrict__ wf,        // 64
                const float* __restrict__ bf,        // 1
                float* __restrict__ pred,            // M
                float* __restrict__ d2f,             // M x 64 x 64 (pre-zeroed)
                int M, int Vv)
{
  __shared__ float sX[16 * STRIDE];      // input tile / backward scratch
  __shared__ float sH[4][16 * STRIDE];   // per-layer activations (ReLU masks)

  const int lane = threadIdx.x;
  const int nl   = lane & 15;   // A-frag: row M; B/C-frag: col N
  const int half = lane >> 4;   // selects K pair (A/B) or M+8 (C)
  const int tile = blockIdx.x * 16;

  // ---- assemble input tile x = [c*u, o0, o1, (volt), f1..f63, pad] ----
  for (int e = lane; e < 16 * STRIDE; e += 32) {
    int m = e / STRIDE;
    int j = e - m * STRIDE;
    int s = tile + m;
    float val = 0.0f;
    if (j < FIN && s < M) {
      int n  = IS_CAP ? (s / Vv) : s;
      int id = idxs[n];
      if (j == 0) {
        float f0 = dk[id * 128] + __expf(0.5f * dk[id * 128 + 64]) * eps[id * 64];
        val = centers[n * 3] * (1e-10f + __expf(-f0));
      } else if (j == 1) {
        val = centers[n * 3 + 1];
      } else if (j == 2) {
        val = centers[n * 3 + 2];
      } else if (IS_CAP && j == 3) {
        val = volts[s % Vv];
      } else {
        int fj = IS_CAP ? (j - 3) : (j - 2);  // feature index 1..63
        val = dk[id * 128 + fj] + __expf(0.5f * dk[id * 128 + 64 + fj]) * eps[id * 64 + fj];
      }
    }
    sX[e] = val;
  }
  __syncthreads();

  // ================= forward =================
  const float* hin = sX;
  for (int l = 0; l < 4; ++l) {
    const float* Wl = (l == 0) ? w0 : (wb + (size_t)(l - 1) * 4096);
    const float* bl = (l == 0) ? b0 : (bb + (l - 1) * 64);

    v8f acc[4] = {};
    // ---- guard-free main K loop (rows 0..63) ----
#pragma unroll
    for (int k = 0; k < 16; ++k) {
      const int kb = 4 * k + 2 * half;
      v2f a = *(const v2f*)(hin + nl * STRIDE + kb);   // ds_load_b64
#pragma unroll
      for (int c = 0; c < 4; ++c) {
        const int col = 16 * c + nl;
        v2f b;
        b.x = Wl[(size_t)(kb)     * 64 + col];
        b.y = Wl[(size_t)(kb + 1) * 64 + col];
        acc[c] = wmma4(a, b, acc[c]);
      }
    }
    // ---- peeled K tail (rows 64..FIN-1), layer 0 only; no divergent loads ----
    if (l == 0 && FIN > 64) {
      const int kb = 64 + 2 * half;
      v2f a = *(const v2f*)(hin + nl * STRIDE + kb);   // LDS zero-padded
#pragma unroll
      for (int c = 0; c < 4; ++c) {
        const int col = 16 * c + nl;
        float t0 = Wl[64 * 64 + col];                          // row 64 (always valid)
        float t1 = (FIN > 65) ? Wl[65 * 64 + col] : 0.0f;      // compile-time
        float t2 = (FIN > 66) ? Wl[66 * 64 + col] : 0.0f;      // compile-time
        v2f b;
        b.x = half ? t2 : t0;      // v_cndmask on values, loads unconditional
        b.y = half ? 0.0f : t1;
        acc[c] = wmma4(a, b, acc[c]);
      }
    }
    // ---- bias + relu + store activations ----
    float* hout = sH[l];
#pragma unroll
    for (int c = 0; c < 4; ++c) {
      const int col  = 16 * c + nl;
      const float bi = bl[col];
#pragma unroll
      for (int r = 0; r < 8; ++r) {              // C: vgpr r -> M = r + 8*half
        const int m = r + 8 * half;
        float v = acc[c][r] + bi;
        hout[m * STRIDE + col] = (v > 0.0f) ? v : 0.0f;
      }
    }
    __syncthreads();
    hin = hout;
  }

  // ---- head: y = h4 . wf + bf (lanes 0..15, one sample each) ----
  float y = 0.0f;
  if (half == 0) {
    float s = bf[0];
    for (int j = 0; j < 64; ++j) s += sH[3][nl * STRIDE + j] * wf[j];
    y = s;
  }

  // ================= backward =================
  // v4 = D4 * wf  -> sX
  for (int e = lane; e < 16 * 64; e += 32) {
    int m = e >> 6, j = e & 63;
    sX[m * STRIDE + j] = (sH[3][m * STRIDE + j] > 0.0f) ? wf[j] : 0.0f;
  }
  __syncthreads();

  // v_l = D_l * (v_{l+1} @ W_l^T), wb idx 2..0
  for (int l = 2; l >= 0; --l) {
    const float* Wl = wb + (size_t)l * 4096;
    v8f acc[4] = {};
#pragma unroll
    for (int k = 0; k < 16; ++k) {
      const int kb = 4 * k + 2 * half;
      v2f a = *(const v2f*)(sX + nl * STRIDE + kb);            // ds_load_b64
#pragma unroll
      for (int c = 0; c < 4; ++c) {
        const int col = 16 * c + nl;
        v2f b = *(const v2f*)(Wl + (size_t)col * 64 + kb);     // global_load_b64
        acc[c] = wmma4(a, b, acc[c]);
      }
    }
    __syncthreads();                             // finish reading sX before overwrite
    const float* hm = sH[l];
#pragma unroll
    for (int c = 0; c < 4; ++c) {
      const int col = 16 * c + nl;
#pragma unroll
      for (int r = 0; r < 8; ++r) {
        const int m = r + 8 * half;
        sX[m * STRIDE + col] = (hm[m * STRIDE + col] > 0.0f) ? acc[c][r] : 0.0f;
      }
    }
    __syncthreads();
  }

  // ---- epilogue: a0 = w0[0,:].v1 ; write pred and the single Hessian entry ----
  if (half == 0) {
    const int s = tile + nl;
    if (s < M) {
      float a0 = 0.0f;
      for (int j = 0; j < 64; ++j) a0 += w0[j] * sX[nl * STRIDE + j];
      const int n  = IS_CAP ? (s / Vv) : s;
      const int id = idxs[n];
      float f0 = dk[id * 128] + __expf(0.5f * dk[id * 128 + 64]) * eps[id * 64];
      float e  = __expf(-f0);
      float u  = 1e-10f + e;
      float cy = centers[n * 3];
      pred[s] = y + a0 * u * (measured[n] - cy);
      d2f[(size_t)s * 4096] = a0 * cy * e;       // d2F[s][0][0]; rest is zero
    }
  }
}

// ---------------------------------------------------------------------------
extern "C" void kernel_launch(void* const* d_in, const int* in_sizes, int n_in,
                              void* d_out, int out_size, void* d_ws, size_t ws_size,
                              hipStream_t stream) {
  (void)n_in; (void)out_size; (void)d_ws; (void)ws_size;

  const float* centers  = (const float*)d_in[0];
  const int*   idxs     = (const int*)  d_in[1];
  const float* measured = (const float*)d_in[2];
  const float* volts    = (const float*)d_in[3];
  const float* eps      = (const float*)d_in[4];
  const float* dk       = (const float*)d_in[5];
  const float* cw0 = (const float*)d_in[6];
  const float* cb0 = (const float*)d_in[7];
  const float* cwb = (const float*)d_in[8];
  const float* cbb = (const float*)d_in[9];
  const float* cwf = (const float*)d_in[10];
  const float* cbf = (const float*)d_in[11];
  const float* vw0 = (const float*)d_in[12];
  const float* vb0 = (const float*)d_in[13];
  const float* vwb = (const float*)d_in[14];
  const float* vbb = (const float*)d_in[15];
  const float* vwf = (const float*)d_in[16];
  const float* vbf = (const float*)d_in[17];

  const int N  = in_sizes[0] / 3;   // 512
  const int V  = in_sizes[3];       // 32
  const int Mc = N * V;             // 16384

  float* out = (float*)d_out;
  const size_t oCap  = 0;
  const size_t oVol  = (size_t)Mc;
  const size_t oMean = oVol + (size_t)N;
  const size_t oLog  = oMean + (size_t)N * 64;
  const size_t oCd2  = oLog  + (size_t)N * 64;
  const size_t oVd2  = oCd2  + (size_t)Mc * 4096;

  // 1) zero-fill cd2F + vd2F (contiguous, 16B aligned) with NT b128 stores
  size_t zcount = ((size_t)Mc + (size_t)N) * 4096;   // floats
  size_t n4 = zcount / 4;
  unsigned int zb = (unsigned int)((n4 + 255) / 256);
  zero_f4_kernel<<<zb, 256, 0, stream>>>((v4f*)(out + oCd2), n4);

  // 2) mean / log_sig gathers
  gather_kernel<<<(N * 64 + 255) / 256, 256, 0, stream>>>(dk, idxs, out + oMean, out + oLog, N * 64);

  // 3) cap network: 16384 samples, 16 per wave
  mlp_hess_kernel<67, true><<<(Mc + 15) / 16, 32, 0, stream>>>(
      centers, idxs, measured, volts, eps, dk,
      cw0, cb0, cwb, cbb, cwf, cbf,
      out + oCap, out + oCd2, Mc, V);

  // 4) vol network: 512 samples
  mlp_hess_kernel<66, false><<<(N + 15) / 16, 32, 0, stream>>>(
      centers, idxs, measured, volts, eps, dk,
      vw0, vb0, vwb, vbb, vwf, vbf,
      out + oVol, out + oVd2, N, 1);
}